// s4_kernel_30081950941872
// MI455X (gfx1250) — compile-verified
//
#include <hip/hip_runtime.h>
#include <math.h>

// S4 Cauchy-kernel generating function at roots of unity, mapped onto the
// CDNA5 FP32 tensor path (V_WMMA_F32_16X16X4_F32) for full fp32 precision.
//
// Per 16-frequency tile (one wave32):
//   D[16x16] += A[16x4] * B[4x16], chained 32x  (K = 128 = 64 poles x {re,im})
//   A[M][2n+0] = Re(1/(g_M - Lam_n)),  A[M][2n+1] = Im(1/(g_M - Lam_n))
//   B[2n+0][2j+0] = Re(v_j[n])   B[2n+0][2j+1] = Im(v_j[n])
//   B[2n+1][2j+0] = -Im(v_j[n])  B[2n+1][2j+1] = Re(v_j[n])
//   v_0=conj(C)*B, v_1=conj(C)*P, v_2=conj(Q)*B, v_3=conj(Q)*P
// so D columns 0..7 are {k00,k01,k10,k11} x {re,im}; columns 8..15 multiply
// zeros. g = i*(2/step)*tan(pi*k/L), c = 1 + i*tan(pi*k/L).
//
// Reciprocals use hardware v_rcp_f32 (~1 ulp) instead of the IEEE division
// expansion: well below this computation's fp32 noise floor and ~2x fewer
// VALU ops per WMMA in the hot loop.

typedef float v2f __attribute__((ext_vector_type(2)));
typedef float v8f __attribute__((ext_vector_type(8)));

#define NSTATE 64
#define WAVES  4
#define TPB    (WAVES * 32)

__global__ __launch_bounds__(TPB)
void s4_cauchy_wmma_f32(const float* __restrict__ Lre, const float* __restrict__ Lim,
                        const float* __restrict__ Pre, const float* __restrict__ Pim,
                        const float* __restrict__ Qre, const float* __restrict__ Qim,
                        const float* __restrict__ Bre, const float* __restrict__ Bim,
                        const float* __restrict__ Cre, const float* __restrict__ Cim,
                        const float* __restrict__ stepPtr,
                        float* __restrict__ out, int L, int numTiles)
{
    __shared__ float2 sLam[NSTATE];            // (Lam_re, Lam_im)
    __shared__ float2 sBV[NSTATE * 16];        // (evenRowVal, oddRowVal) per (n, col)
    __shared__ float  sDump[WAVES][16 * 16];   // per-wave D-matrix spill

    const int tid = threadIdx.x;

    // ---- build the Lambda table ----
    if (tid < NSTATE) sLam[tid] = make_float2(Lre[tid], Lim[tid]);

    // ---- build the B-matrix constant table (cols 8..15 are zero) ----
    for (int idx = tid; idx < NSTATE * 16; idx += TPB) {
        const int n = idx >> 4, col = idx & 15;
        float ev = 0.0f, od = 0.0f;
        if (col < 8) {
            const int j = col >> 1;      // combo: 0=a0b0 1=a0b1 2=a1b0 3=a1b1
            const int p = col & 1;       // 0 = re column, 1 = im column
            float are, aim, bre, bim;
            if (j < 2) { are = Cre[n]; aim = -Cim[n]; }   // conj(C)
            else       { are = Qre[n]; aim = -Qim[n]; }   // conj(Q)
            if ((j & 1) == 0) { bre = Bre[n]; bim = Bim[n]; }
            else              { bre = Pre[n]; bim = Pim[n]; }
            const float vre = are * bre - aim * bim;
            const float vim = are * bim + aim * bre;
            if (p == 0) { ev = vre;  od = -vim; }         // contributes to k_re
            else        { ev = vim;  od =  vre; }         // contributes to k_im
        }
        sBV[idx] = make_float2(ev, od);
    }
    __syncthreads();

    const int wave = tid >> 5;
    const int lane = tid & 31;
    const int tile = blockIdx.x * WAVES + wave;
    if (tile >= numTiles) return;                 // wave-uniform guard

    const int col = lane & 15;                    // A: row M   | B: column N
    const int hi  = lane >> 4;

    // per-lane frequency (for the A side and the epilogue)
    const int   kf   = tile * 16 + col;
    const float step = stepPtr[0];
    const float PIf  = 3.14159265358979323846f;
    const float t    = tanf(PIf * (float)kf / (float)L);  // tan(theta/2)
    const float g_im = (2.0f / step) * t;                 // g = i*g_im

    v8f acc = {};
    #pragma unroll 4
    for (int q = 0; q < 32; ++q) {
        const int n = 2 * q + hi;                 // pole index handled by this lane
        const float2 lam = sLam[n];
        const float dre = -lam.x;                 // Re(g - Lam) = -Lam_re (> 0)
        const float dim = g_im - lam.y;           // Im(g - Lam)
        const float inv = __builtin_amdgcn_rcpf(dre * dre + dim * dim); // v_rcp_f32
        v2f a; a.x = dre * inv; a.y = -dim * inv; // w = conj(d)/|d|^2
        const float2 bv = sBV[(n << 4) + col];
        v2f b; b.x = bv.x; b.y = bv.y;
        // D = A*B + C, FP32 tensor op (EXEC all-ones here)
        acc = __builtin_amdgcn_wmma_f32_16x16x4_f32(
                  false, a, false, b, (short)0, acc, false, false);
    }

    // ---- spill D to LDS: lane holds D[M = r + 8*hi][N = col] in acc[r] ----
    float* dump = sDump[wave];
    #pragma unroll
    for (int r = 0; r < 8; ++r)
        dump[(r + 8 * hi) * 16 + col] = acc[r];
    __builtin_amdgcn_wave_barrier();              // same-wave DS ops are in-order

    // ---- final complex combine: out = c * (k00 - k01*k10/(1+k11)) ----
    if (lane < 16) {
        const int M = lane;                       // frequency row; t matches (col==lane)
        const float k00r = dump[M * 16 + 0], k00i = dump[M * 16 + 1];
        const float k01r = dump[M * 16 + 2], k01i = dump[M * 16 + 3];
        const float k10r = dump[M * 16 + 4], k10i = dump[M * 16 + 5];
        const float k11r = dump[M * 16 + 6], k11i = dump[M * 16 + 7];

        const float rr = 1.0f + k11r, ri = k11i;
        const float rinv = __builtin_amdgcn_rcpf(rr * rr + ri * ri);
        const float ir =  rr * rinv, ii = -ri * rinv;     // 1/(1+k11)
        const float pr = k01r * k10r - k01i * k10i;       // k01*k10
        const float pi = k01r * k10i + k01i * k10r;
        const float qr = pr * ir - pi * ii;               // * 1/(1+k11)
        const float qi = pr * ii + pi * ir;
        const float sr = k00r - qr, si = k00i - qi;
        // c = 1 + i*t  ->  (sr - t*si) + i*(si + t*sr)
        const float outr = sr - t * si;
        const float outi = si + t * sr;

        const int k = tile * 16 + M;
        if (k < L)
            reinterpret_cast<float2*>(out)[k] = make_float2(outr, outi);
    }
}

extern "C" void kernel_launch(void* const* d_in, const int* in_sizes, int n_in,
                              void* d_out, int out_size, void* d_ws, size_t ws_size,
                              hipStream_t stream) {
    (void)in_sizes; (void)n_in; (void)d_ws; (void)ws_size;
    const float* Lre = (const float*)d_in[0];
    const float* Lim = (const float*)d_in[1];
    const float* Pre = (const float*)d_in[2];
    const float* Pim = (const float*)d_in[3];
    const float* Qre = (const float*)d_in[4];
    const float* Qim = (const float*)d_in[5];
    const float* Bre = (const float*)d_in[6];
    const float* Bim = (const float*)d_in[7];
    const float* Cre = (const float*)d_in[8];
    const float* Cim = (const float*)d_in[9];
    const float* stp = (const float*)d_in[10];

    const int L        = out_size / 2;            // complex64 flattened to floats
    const int numTiles = (L + 15) / 16;
    const int blocks   = (numTiles + WAVES - 1) / WAVES;

    s4_cauchy_wmma_f32<<<blocks, TPB, 0, stream>>>(
        Lre, Lim, Pre, Pim, Qre, Qim, Bre, Bim, Cre, Cim, stp,
        (float*)d_out, L, numTiles);
}